// LinkClassifier_6201932775771
// MI455X (gfx1250) — compile-verified
//
#include <hip/hip_runtime.h>
#include <hip/hip_bf16.h>

// Problem constants (match reference)
#define N_NODES 8192
#define D_IN    256
#define D_EMB   256
#define NE      65536
// P = N*(N-1)/2 = 33550336

typedef __attribute__((ext_vector_type(2))) float v2f;
typedef __attribute__((ext_vector_type(8))) float v8f;

// Workspace layout (float elements):
//   [0      .. 4095 ]  Wab[256][16]  cols 0..1 = W_enc@W_cls[:D] (A-proj),
//                                    cols 2..3 = W_enc@W_cls[D:] (B-proj), cols 4..15 = 0
//   [4096   .. 4097 ]  cbias[2]      b_enc@(Wc1+Wc2) + b_cls
//   [4112   .. +16383] A[N][2]
//   [A+16384.. +16383] B[N][2]
#define WS_WAB   0
#define WS_CBIAS 4096
#define WS_A     4112
#define WS_B     (WS_A + 2 * N_NODES)

// ---------------------------------------------------------------------------
// Kernel 1: fuse encoder + classifier weights.
// Wab[d][c] = sum_k W_enc[d,k] * W_cls[k + (c>=2 ? 256 : 0), c&1]   (c < 4)
// cbias[c]  = b_cls[c] + sum_k b_enc[k]*(W_cls[k,c] + W_cls[256+k,c])
// 0.26 MFLOP total -> scalar is fine.
// ---------------------------------------------------------------------------
__global__ __launch_bounds__(256)
void fuse_weights(const float* __restrict__ W_enc, const float* __restrict__ b_enc,
                  const float* __restrict__ W_cls, const float* __restrict__ b_cls,
                  float* __restrict__ ws) {
    const int c = blockIdx.x;    // 0..15 (column of padded Wab)
    const int d = threadIdx.x;   // 0..255 (row)
    float acc = 0.0f;
    if (c < 4) {
        const int col  = c & 1;
        const int half = c >> 1;                       // 0 -> A-proj, 1 -> B-proj
        const float* wc = W_cls + (size_t)half * D_EMB * 2;
        const float* we = W_enc + (size_t)d * D_EMB;
        for (int k = 0; k < D_EMB; ++k)
            acc += we[k] * wc[k * 2 + col];
    }
    ws[WS_WAB + d * 16 + c] = acc;                     // zero-pad cols 4..15
    if (c == 15 && d < 2) {
        float cb = b_cls[d];
        for (int k = 0; k < D_EMB; ++k)
            cb += b_enc[k] * (W_cls[k * 2 + d] + W_cls[(D_EMB + k) * 2 + d]);
        ws[WS_CBIAS + d] = cb;
    }
}

// ---------------------------------------------------------------------------
// Kernel 2: A/B projections via full-fp32 WMMA.
// One wave32 per 16-row tile; C(16x16) = nodes_tile(16x256) @ Wab(256x16)
// accumulated with 64x V_WMMA_F32_16X16X4_F32 (K-step 4, f32 end-to-end).
// A-operand layout (ISA 7.12.2, 32-bit A 16x4): lane = M (mod 16),
//   lanes 0-15 hold K={0,1}, lanes 16-31 hold K={2,3}.
// B-operand (4x16): rows striped across lanes within a VGPR, same half split.
// C/D layout: VGPR v -> row v (lanes 0-15) / row v+8 (lanes 16-31), col = lane&15.
// ---------------------------------------------------------------------------
__global__ __launch_bounds__(32)
void encode_ab(const float* __restrict__ nodes, float* ws) {
    const int lane = threadIdx.x;
    const int half = lane >> 4;      // 0: K offset 0, 1: K offset 2
    const int m    = lane & 15;      // row within tile / column of B,C
    const int row0 = blockIdx.x * 16;

    const float* Wab  = ws + WS_WAB;
    const float* arow = nodes + (size_t)(row0 + m) * D_IN + half * 2;

    v8f c = {};   // C starts at 0 (bias folded into cbias at pair time)
#pragma unroll 8
    for (int kk = 0; kk < D_IN; kk += 4) {
        v2f a, b;
        a.x = arow[kk];
        a.y = arow[kk + 1];
        const int kb = kk + half * 2;
        b.x = Wab[(kb    ) * 16 + m];
        b.y = Wab[(kb + 1) * 16 + m];
        // (neg_a, A, neg_b, B, c_mod, C, reuse_a, reuse_b)
        c = __builtin_amdgcn_wmma_f32_16x16x4_f32(false, a, false, b,
                                                  (short)0, c, false, false);
    }

    // Scatter C columns 0..3 into compact A/B arrays.
    float* Aout = ws + WS_A;
    float* Bout = ws + WS_B;
#pragma unroll
    for (int v = 0; v < 8; ++v) {
        const int row = row0 + v + half * 8;
        if (m < 2)       Aout[row * 2 + m]       = c[v];
        else if (m < 4)  Bout[row * 2 + (m - 2)] = c[v];
    }
}

// ---------------------------------------------------------------------------
// Kernel 3: the bandwidth-bound part. For every pair (s,t), s<t:
//   y[p] = A[s] + B[t] + cbias ;  y_true[p] = 0
// p = s*(2N-s-1)/2 + (t-s-1)  (matches jnp.triu_indices row-major order).
// 403 MB of streaming writes -> non-temporal stores (stream > 192 MB L2).
// Grid: blockIdx.y = s, blockIdx.x covers t in 256-wide slabs.
// ---------------------------------------------------------------------------
__global__ __launch_bounds__(256)
void pair_logits(const float* __restrict__ ws, float* __restrict__ y,
                 int* __restrict__ y_true) {
    const int s = blockIdx.y;
    const int t = s + 1 + blockIdx.x * 256 + threadIdx.x;
    if (t >= N_NODES) return;

    const v2f* A = (const v2f*)(ws + WS_A);
    const v2f* B = (const v2f*)(ws + WS_B);
    const v2f as = A[s];           // uniform across block -> broadcast from cache
    const v2f bt = B[t];           // coalesced 8B per lane
    const float cb0 = ws[WS_CBIAS + 0];
    const float cb1 = ws[WS_CBIAS + 1];

    const unsigned us = (unsigned)s;
    const size_t p = (size_t)(us * (2u * N_NODES - us - 1u) / 2u
                              + (unsigned)(t - s - 1));

    v2f out;
    out.x = as.x + bt.x + cb0;
    out.y = as.y + bt.y + cb1;
    __builtin_nontemporal_store(out, (v2f*)y + p);   // 8B NT store
    __builtin_nontemporal_store(0,   y_true + p);    // zero-fill labels
}

// ---------------------------------------------------------------------------
// Kernel 4: scatter edge-membership labels (after zero-fill, stream-ordered).
// Value is always 1 -> plain stores, duplicates/self-conflicts harmless.
// ---------------------------------------------------------------------------
__global__ __launch_bounds__(256)
void edge_labels(const int* __restrict__ edges, int* __restrict__ y_true) {
    const int i = blockIdx.x * 256 + threadIdx.x;
    if (i >= NE) return;
    const unsigned u = (unsigned)edges[i];
    const unsigned v = (unsigned)edges[NE + i];
    if (u < v) {
        const unsigned p = u * (2u * N_NODES - u - 1u) / 2u + (v - u - 1u);
        y_true[p] = 1;
    }
}

// ---------------------------------------------------------------------------
extern "C" void kernel_launch(void* const* d_in, const int* in_sizes, int n_in,
                              void* d_out, int out_size, void* d_ws, size_t ws_size,
                              hipStream_t stream) {
    const float* nodes = (const float*)d_in[0];   // [8192, 256] f32
    const int*   edges = (const int*)  d_in[1];   // [2, 65536] i32
    const float* W_enc = (const float*)d_in[2];   // [256, 256] f32
    const float* b_enc = (const float*)d_in[3];   // [256] f32
    const float* W_cls = (const float*)d_in[4];   // [512, 2] f32
    const float* b_cls = (const float*)d_in[5];   // [2] f32

    float* ws = (float*)d_ws;
    const size_t P = (size_t)N_NODES * (N_NODES - 1) / 2;
    float* y      = (float*)d_out;                 // [P, 2] f32
    int*   y_true = (int*)((float*)d_out + 2 * P); // [P] i32 (bit layout matches)

    fuse_weights<<<dim3(16), dim3(256), 0, stream>>>(W_enc, b_enc, W_cls, b_cls, ws);
    encode_ab  <<<dim3(N_NODES / 16), dim3(32), 0, stream>>>(nodes, ws);
    pair_logits<<<dim3(32, N_NODES), dim3(256), 0, stream>>>(ws, y, y_true);
    edge_labels<<<dim3(NE / 256), dim3(256), 0, stream>>>(edges, y_true);
}